// DQN_2585570312620
// MI455X (gfx1250) — compile-verified
//
#include <hip/hip_runtime.h>

// Problem constants (from reference)
#define BB    16
#define NN    5000
#define EE    80000
#define FEAT  50
#define HID   128
#define PADR  1000

typedef __attribute__((ext_vector_type(2))) float v2f;
typedef __attribute__((ext_vector_type(8))) float v8f;

__device__ __forceinline__ v8f wmma_f32x4(v2f a, v2f b, v8f c) {
  // V_WMMA_F32_16X16X4_F32 : D(16x16 f32) = A(16x4 f32) x B(4x16 f32) + C
  return __builtin_amdgcn_wmma_f32_16x16x4_f32(
      /*neg_a=*/false, a, /*neg_b=*/false, b,
      /*c_mod=*/(short)0, c, /*reuse_a=*/false, /*reuse_b=*/false);
}

__device__ __forceinline__ int imin(int a, int b) { return a < b ? a : b; }

// ---------------------------------------------------------------- utilities

__global__ void fill_zero_kernel(float* __restrict__ p, long n) {
  long i = (long)blockIdx.x * blockDim.x + threadIdx.x;
  long stride = (long)gridDim.x * blockDim.x;
  for (; i < n; i += stride) p[i] = 0.0f;
}

// cnt[b*N + dst] += 1 over all edges
__global__ void count_edges_kernel(const int* __restrict__ ei, float* __restrict__ cnt) {
  long total = (long)BB * EE;
  long i = (long)blockIdx.x * blockDim.x + threadIdx.x;
  long stride = (long)gridDim.x * blockDim.x;
  for (; i < total; i += stride) {
    int e = (int)(i % EE);
    int b = (int)(i / EE);
    int dst = ei[(long)b * 2 * EE + EE + e];
    atomicAdd(&cnt[(long)b * NN + dst], 1.0f);
  }
}

// agg[b, dst, f] += x[b, src, f]   (f fastest-varying -> coalesced)
__global__ void scatter_add_kernel(const float* __restrict__ x, const int* __restrict__ ei,
                                   float* __restrict__ agg, int F) {
  long total = (long)BB * EE * F;
  long i = (long)blockIdx.x * blockDim.x + threadIdx.x;
  long stride = (long)gridDim.x * blockDim.x;
  for (; i < total; i += stride) {
    int f = (int)(i % F);
    long t = i / F;
    int e = (int)(t % EE);
    int b = (int)(t / EE);
    const int* eib = ei + (long)b * 2 * EE;
    int src = eib[e];
    int dst = eib[EE + e];
    atomicAdd(&agg[((long)b * NN + dst) * F + f], x[((long)b * NN + src) * F + f]);
  }
}

// agg[row, f] /= max(cnt[row], 1)
__global__ void mean_div_kernel(float* __restrict__ agg, const float* __restrict__ cnt, int F) {
  long total = (long)BB * NN * F;
  long i = (long)blockIdx.x * blockDim.x + threadIdx.x;
  long stride = (long)gridDim.x * blockDim.x;
  for (; i < total; i += stride) {
    float c = cnt[i / F];
    agg[i] /= fmaxf(c, 1.0f);
  }
}

// ---------------------------------------------------------------- WMMA GEMM
// C[M x NCT] = act( A@Wa + (A2 ? A2@Wb : 0) + bias ),  NCT compile-time.
// One wave computes a 16-row x (NT*16)-col strip (NT accumulator tiles).
// Hot loop per K-step: one global_load_b64 (A fragment), 2 cndmasks,
// 2*NT global_load_b32 off ONE address register with immediate offsets,
// NT back-to-back v_wmma_f32_16x16x4_f32, and two pointer increments.
// pad_mode=1: output row r -> b=r/pad_rows, rr=r%pad_rows; A row is
//             b*src_rows+rr, reading zero when rr >= *act_range_p.
template <int NT, int NCT>
__global__ void wmma_gemm_kernel(const float* __restrict__ A,
                                 const float* __restrict__ A2,
                                 const float* __restrict__ Wa,
                                 const float* __restrict__ Wb,
                                 const float* __restrict__ bias,
                                 float* __restrict__ C,
                                 int M, int K, int relu,
                                 int pad_mode, const int* __restrict__ act_range_p,
                                 int pad_rows, int src_rows) {
  const int lane = threadIdx.x & 31;
  const int mt = blockIdx.x * (blockDim.x >> 5) + (threadIdx.x >> 5);
  const int tm = (M + 15) >> 4;
  if (mt >= tm) return;                 // wave-uniform: EXEC stays all-ones

  const int ml = lane & 15;             // M (A) / N (B,C) position
  const int g  = lane >> 4;             // K half-group: VGPR j, group g -> K = 2g + j

  const int row = mt * 16 + ml;
  bool rvalid = row < M;
  long arow;
  if (pad_mode) {
    const int rc = imin(row, M - 1);
    const int b  = rc / pad_rows;
    const int rr = rc - b * pad_rows;
    rvalid = rvalid && (rr < *act_range_p);
    arow = (long)b * src_rows + rr;     // rr < pad_rows <= src_rows: load safe
  } else {
    arow = imin(row, M - 1);            // clamped: load safe
  }

  v8f acc[NT];
#pragma unroll
  for (int t = 0; t < NT; ++t) acc[t] = (v8f){};

  const int npass = (A2 != nullptr) ? 2 : 1;

  for (int pass = 0; pass < npass; ++pass) {
    const float* __restrict__ Ap = pass ? A2 : A;
    const float* __restrict__ Wp = pass ? Wb : Wa;

    const float* aptr = Ap + arow * (long)K + 2 * g;   // even -> 8B aligned
    const float* bptr = Wp + (2 * g) * NCT + ml;
    const int nfull = K >> 2;

    for (int it = 0; it < nfull; ++it) {
      v2f af = *(const v2f*)aptr;
      af.x = rvalid ? af.x : 0.0f;
      af.y = rvalid ? af.y : 0.0f;
#pragma unroll
      for (int t = 0; t < NT; ++t) {
        v2f bf;
        bf.x = bptr[t * 16];            // immediate offset: t*64 bytes
        bf.y = bptr[NCT + t * 16];      // immediate offset: (NCT+t*16)*4 bytes
        acc[t] = wmma_f32x4(af, bf, acc[t]);
      }
      aptr += 4;
      bptr += 4 * NCT;
    }

    // tail (K % 4 != 0): clamped scalar loads + selects, no exec branches
    const int k0 = nfull << 2;
    if (k0 < K) {
      const float* __restrict__ Arow_p = Ap + arow * (long)K;
      const int ka  = k0 + 2 * g;
      const int kc0 = imin(ka, K - 1);
      const int kc1 = imin(ka + 1, K - 1);
      float a0 = Arow_p[kc0];
      float a1 = Arow_p[kc1];
      v2f af;
      af.x = (rvalid && ka     < K) ? a0 : 0.0f;
      af.y = (rvalid && ka + 1 < K) ? a1 : 0.0f;
      const float* __restrict__ W0 = Wp + (long)kc0 * NCT + ml;
      const float* __restrict__ W1 = Wp + (long)kc1 * NCT + ml;
#pragma unroll
      for (int t = 0; t < NT; ++t) {
        v2f bf;
        bf.x = W0[t * 16];
        bf.y = W1[t * 16];
        acc[t] = wmma_f32x4(af, bf, acc[t]);
      }
    }
  }

  // epilogue: bias + activation + stores with immediate offsets
  float* crow = C + (long)(mt * 16 + (g << 3)) * NCT + ml;
#pragma unroll
  for (int t = 0; t < NT; ++t) {
    const int n = t * 16 + ml;
    const bool nvalid = (NT * 16 == NCT) ? true : (n < NCT);
    const float bv = (bias != nullptr) ? bias[imin(n, NCT - 1)] : 0.0f;
#pragma unroll
    for (int v = 0; v < 8; ++v) {
      const int r = mt * 16 + v + (g << 3);
      float val = acc[t][v] + bv;
      if (relu) val = fmaxf(val, 0.0f);
      if (r < M && nvalid) crow[v * NCT + t * 16] = val;  // imm offset store
    }
  }
}

// ---------------------------------------------------------------- final 128->1
__global__ void final_dot_kernel(const float* __restrict__ y, const float* __restrict__ W3,
                                 const float* __restrict__ b3, float* __restrict__ out, int M) {
  const int lane = threadIdx.x & 31;
  const int row = blockIdx.x * (blockDim.x >> 5) + (threadIdx.x >> 5);
  if (row >= M) return;
  float s = 0.0f;
#pragma unroll
  for (int k = lane; k < HID; k += 32) s += y[(long)row * HID + k] * W3[k];
#pragma unroll
  for (int off = 16; off > 0; off >>= 1) s += __shfl_xor(s, off, 32);
  if (lane == 0) out[row] = s + b3[0];
}

// ---------------------------------------------------------------- launch

static inline int strip_grid(int M) {
  int tm = (M + 15) / 16;
  return (tm + 3) / 4;   // 4 waves (16-row strips) per 128-thread block
}

extern "C" void kernel_launch(void* const* d_in, const int* in_sizes, int n_in,
                              void* d_out, int out_size, void* d_ws, size_t ws_size,
                              hipStream_t stream) {
  (void)in_sizes; (void)n_in; (void)out_size; (void)ws_size;

  const float* edge_feat = (const float*)d_in[0];   // B x N x FEAT
  const int*   edge_idx  = (const int*)d_in[1];     // B x 2 x E
  const int*   act_range = (const int*)d_in[2];     // scalar (800)
  const float* Wl1 = (const float*)d_in[3];         // FEAT x HID
  const float* Wr1 = (const float*)d_in[4];         // FEAT x HID
  const float* b1c = (const float*)d_in[5];         // HID
  const float* Wl2 = (const float*)d_in[6];         // HID x FEAT
  const float* Wr2 = (const float*)d_in[7];         // HID x FEAT
  const float* b2c = (const float*)d_in[8];         // FEAT
  const float* W1  = (const float*)d_in[9];         // FEAT x 128
  const float* b1  = (const float*)d_in[10];        // 128
  const float* W2  = (const float*)d_in[11];        // 128 x 128
  const float* b2  = (const float*)d_in[12];        // 128
  const float* W3  = (const float*)d_in[13];        // 128 x 1
  const float* b3  = (const float*)d_in[14];        // 1
  float* out = (float*)d_out;                       // B x PADR

  // workspace layout (floats)
  const long S_AGG1 = (long)BB * NN * FEAT;   // 4,000,000  (agg1 / mean1 / x2)
  const long S_CNT  = (long)BB * NN;          //    80,000
  const long S_H    = (long)BB * NN * HID;    // 10,240,000
  const long S_Y    = (long)BB * PADR * HID;  //  2,048,000

  float* agg1 = (float*)d_ws;
  float* cnt  = agg1 + S_AGG1;
  float* h    = cnt  + S_CNT;
  float* agg2 = h    + S_H;
  float* y1   = agg2 + S_H;
  float* y2   = y1   + S_Y;

  const int TB = 256;
  const long bigE1 = (long)BB * EE * FEAT;
  const long bigE2 = (long)BB * EE * HID;

  // ---- SAGE layer 1 ----
  fill_zero_kernel<<<(int)((S_AGG1 + S_CNT + TB - 1) / TB), TB, 0, stream>>>(agg1, S_AGG1 + S_CNT);
  count_edges_kernel<<<(int)(((long)BB * EE + TB - 1) / TB), TB, 0, stream>>>(edge_idx, cnt);
  scatter_add_kernel<<<(int)((bigE1 + TB - 1) / TB), TB, 0, stream>>>(edge_feat, edge_idx, agg1, FEAT);
  mean_div_kernel<<<(int)((S_AGG1 + TB - 1) / TB), TB, 0, stream>>>(agg1, cnt, FEAT);
  // h = relu(mean1 @ Wl1 + x @ Wr1 + b1c)   [M=B*N, K=FEAT, NC=HID] -> <8,128>
  wmma_gemm_kernel<8, 128><<<strip_grid(BB * NN), 128, 0, stream>>>(
      agg1, edge_feat, Wl1, Wr1, b1c, h, BB * NN, FEAT, /*relu=*/1,
      /*pad_mode=*/0, nullptr, 0, 0);

  // ---- SAGE layer 2 ----
  fill_zero_kernel<<<(int)((S_H + TB - 1) / TB), TB, 0, stream>>>(agg2, S_H);
  scatter_add_kernel<<<(int)((bigE2 + TB - 1) / TB), TB, 0, stream>>>(h, edge_idx, agg2, HID);
  mean_div_kernel<<<(int)((S_H + TB - 1) / TB), TB, 0, stream>>>(agg2, cnt, HID);
  // x2 = mean2 @ Wl2 + h @ Wr2 + b2c   [M=B*N, K=HID, NC=FEAT] -> <4,50>, reuse agg1
  wmma_gemm_kernel<4, 50><<<strip_grid(BB * NN), 128, 0, stream>>>(
      agg2, h, Wl2, Wr2, b2c, agg1, BB * NN, HID, /*relu=*/0,
      /*pad_mode=*/0, nullptr, 0, 0);

  // ---- MLP on first action_range rows, zero-padded to PADR ----
  // y1 = relu(xpad @ W1 + b1)   [M=B*PADR, K=FEAT, NC=128] -> <8,128>
  wmma_gemm_kernel<8, 128><<<strip_grid(BB * PADR), 128, 0, stream>>>(
      agg1, nullptr, W1, nullptr, b1, y1, BB * PADR, FEAT, /*relu=*/1,
      /*pad_mode=*/1, act_range, PADR, NN);
  // y2 = relu(y1 @ W2 + b2)   [M=B*PADR, K=128, NC=128] -> <8,128>
  wmma_gemm_kernel<8, 128><<<strip_grid(BB * PADR), 128, 0, stream>>>(
      y1, nullptr, W2, nullptr, b2, y2, BB * PADR, HID, /*relu=*/1,
      /*pad_mode=*/0, nullptr, 0, 0);
  // out = y2 @ W3 + b3   [M=B*PADR, K=128, NC=1]
  final_dot_kernel<<<(BB * PADR + 3) / 4, 128, 0, stream>>>(y2, W3, b3, out, BB * PADR);
}